// nucleusSampling_76622216560925
// MI455X (gfx1250) — compile-verified
//
#include <hip/hip_runtime.h>
#include <stdint.h>

// Nucleus (top-p) sampling + greedy pick of surviving logits.
// Key identity: the top-1 logit always survives the nucleus filter (the
// remove-mask is shifted right by one), so max/argmax of the filtered logits
// == max/argmax of the raw logits. The kernel is therefore a pure
// bandwidth-bound row-wise fused max+argmax: 512 x 50257 f32 = ~103 MB read,
// floor ~4.4 us at 23.3 TB/s.
//
// CDNA5 path: GLOBAL_LOAD_ASYNC_TO_LDS_B128 (ASYNCcnt data mover) in a 4-deep
// software pipeline of 8 KB chunks (2 x b128 per lane per chunk), giving
// ~16 MB in flight across 512 workgroups (~ the 23.3 TB/s x 600 ns
// bandwidth-delay product). s_wait_asynccnt for in-order completion, wave32
// shuffle butterfly for the reduction. WMMA intentionally unused: no matmul.

#define COLS          50257
#define TPB           256
#define WAVES         (TPB / 32)
#define LANE_FLOATS   8                          // 2 x float4 per lane per chunk
#define CHUNK_FLOATS  (TPB * LANE_FLOATS)        // 2048 floats = 8192 B
#define CHUNK_BYTES   (CHUNK_FLOATS * 4)
#define DEPTH         4                          // chunks in flight (32 KB LDS)

// One 16-byte async global->LDS copy (GVS mode: SGPR base + VGPR byte offset).
// VDST operand carries the LDS byte address in a VGPR.
#define ASYNC_LOAD_B128(ldsoff, goff, sbase)                                   \
    asm volatile("global_load_async_to_lds_b128 %0, %1, %2"                    \
                 :: "v"(ldsoff), "v"(goff), "s"(sbase) : "memory")

#define S_WAIT_ASYNCCNT(lit) asm volatile("s_wait_asynccnt " lit ::: "memory")

__device__ __forceinline__ void upd(float v, int i, float& bv, int& bi) {
    if (v > bv) { bv = v; bi = i; }   // per-lane indices ascend -> first-occurrence
}

__global__ __launch_bounds__(TPB)
void nucleus_rowmax_kernel(const float* __restrict__ logits,
                           float* __restrict__ out, int rows) {
    __shared__ alignas(16) float stage[DEPTH][CHUNK_FLOATS];  // 32 KB staging
    __shared__ float swv[WAVES];
    __shared__ int   swi[WAVES];

    const int row = blockIdx.x;
    const int tid = threadIdx.x;
    const float* __restrict__ rowp = logits + (size_t)row * COLS;

    float bv = -__builtin_inff();
    int   bi = 0;

    // ---- alignment prologue: COLS % 4 == 1, so the row base is misaligned by
    // (row % 4) floats. Consume `pre` leading floats scalar so every async
    // b128 below hits a 16-byte-aligned global address.
    const int pre = (4 - (row & 3)) & 3;
    if (tid < pre) upd(rowp[tid], tid, bv, bi);

    const int M    = COLS - pre;             // aligned region (floats)
    const int nch  = M / CHUNK_FLOATS;       // full 8 KB chunks (24 here)
    const int tail = M % CHUNK_FLOATS;

    // Generic pointer to __shared__ carries the LDS byte offset in its low 32
    // bits (LDS aperture); lane slice = 32 bytes.
    const uint32_t ldsBase = (uint32_t)(uintptr_t)(&stage[0][0]) + (uint32_t)(tid * 32);
    const uint32_t gBase   = (uint32_t)(pre * 4 + tid * 32);  // byte offset in row

    // ---- fill: issue chunks 0 .. min(DEPTH,nch)-1 (2 async b128 per lane each)
    {
        const int npre = (nch < DEPTH) ? nch : DEPTH;
        for (int c = 0; c < npre; ++c) {
            const uint32_t l = ldsBase + (uint32_t)(c & (DEPTH - 1)) * CHUNK_BYTES;
            const uint32_t g = gBase + (uint32_t)c * CHUNK_BYTES;
            ASYNC_LOAD_B128(l,      g,      rowp);
            ASYNC_LOAD_B128(l + 16, g + 16, rowp);
        }
    }

    int c = 0;
    // ---- steady state. Async loads complete in order per wave; with
    // 2*(c+DEPTH) issued, asynccnt <= 2*DEPTH-2 (=6) implies both loads of
    // chunk c have landed in LDS. Each lane reads back only its own slice,
    // so its own wave's counter is the only dependency (no barrier needed).
    for (; c + DEPTH < nch; ++c) {
        S_WAIT_ASYNCCNT("6");
        const float* sp = &stage[c & (DEPTH - 1)][tid * LANE_FLOATS];
        const float4 a = *reinterpret_cast<const float4*>(sp);
        const float4 b = *reinterpret_cast<const float4*>(sp + 4);
        const int ib = pre + c * CHUNK_FLOATS + tid * LANE_FLOATS;
        upd(a.x, ib + 0, bv, bi); upd(a.y, ib + 1, bv, bi);
        upd(a.z, ib + 2, bv, bi); upd(a.w, ib + 3, bv, bi);
        upd(b.x, ib + 4, bv, bi); upd(b.y, ib + 5, bv, bi);
        upd(b.z, ib + 6, bv, bi); upd(b.w, ib + 7, bv, bi);
        const uint32_t l = ldsBase + (uint32_t)((c + DEPTH) & (DEPTH - 1)) * CHUNK_BYTES;
        const uint32_t g = gBase + (uint32_t)(c + DEPTH) * CHUNK_BYTES;
        ASYNC_LOAD_B128(l,      g,      rowp);
        ASYNC_LOAD_B128(l + 16, g + 16, rowp);
    }
    // ---- drain: everything issued; wait for all, consume the rest.
    S_WAIT_ASYNCCNT("0");
    for (; c < nch; ++c) {
        const float* sp = &stage[c & (DEPTH - 1)][tid * LANE_FLOATS];
        const float4 a = *reinterpret_cast<const float4*>(sp);
        const float4 b = *reinterpret_cast<const float4*>(sp + 4);
        const int ib = pre + c * CHUNK_FLOATS + tid * LANE_FLOATS;
        upd(a.x, ib + 0, bv, bi); upd(a.y, ib + 1, bv, bi);
        upd(a.z, ib + 2, bv, bi); upd(a.w, ib + 3, bv, bi);
        upd(b.x, ib + 4, bv, bi); upd(b.y, ib + 5, bv, bi);
        upd(b.z, ib + 6, bv, bi); upd(b.w, ib + 7, bv, bi);
    }

    // ---- tail: scalar global loads (per-lane indices stay ascending).
    const int tbase = pre + nch * CHUNK_FLOATS;
    for (int i = tid; i < tail; i += TPB)
        upd(rowp[tbase + i], tbase + i, bv, bi);

    // ---- wave32 butterfly on (val, idx), tie-break smaller index.
    #pragma unroll
    for (int off = 16; off > 0; off >>= 1) {
        float ov = __shfl_xor(bv, off, 32);
        int   oi = __shfl_xor(bi, off, 32);
        if (ov > bv || (ov == bv && oi < bi)) { bv = ov; bi = oi; }
    }

    const int lane = tid & 31;
    const int wave = tid >> 5;
    if (lane == 0) { swv[wave] = bv; swi[wave] = bi; }
    __syncthreads();

    if (tid == 0) {
        for (int w = 1; w < WAVES; ++w) {
            if (swv[w] > bv || (swv[w] == bv && swi[w] < bi)) {
                bv = swv[w]; bi = swi[w];
            }
        }
        // Tuple output, concatenated flat: [tokens (as float) | scores].
        out[row]        = (float)bi;   // token index (exact in f32: < 2^24)
        out[rows + row] = bv;          // surviving-max logit
    }
}

extern "C" void kernel_launch(void* const* d_in, const int* in_sizes, int n_in,
                              void* d_out, int out_size, void* d_ws, size_t ws_size,
                              hipStream_t stream) {
    (void)n_in; (void)out_size; (void)d_ws; (void)ws_size;
    const float* logits = (const float*)d_in[0];
    float* out = (float*)d_out;
    const int rows = in_sizes[0] / COLS;   // 512 for the reference shapes
    nucleus_rowmax_kernel<<<rows, TPB, 0, stream>>>(logits, out, rows);
}